// RNATertiaryModel_8727373545933
// MI455X (gfx1250) — compile-verified
//
#include <hip/hip_runtime.h>
#include <math.h>

typedef __attribute__((ext_vector_type(2))) float v2f;
typedef __attribute__((ext_vector_type(4))) float v4f;
typedef __attribute__((ext_vector_type(8))) float v8f;

#define BATCH 2
#define SEQL  1536
#define NTILE (SEQL / 16)

// Workspace layout (float offsets), all 16-float aligned:
//  Rv12 [B][L][12] : vec(R_new) row-major, padded 9->12 with zeros
//  Rc   [B][3][L][4] : column c of R_new (R[a][c], a=0..2), padded with zero
//  Tp   [B][L][4] : t_new padded with zero
//  q    [B][L]    : |t_new|^2
//  s    [B][L][3] : (R_newT t_new)_c
#define RV12_OFF 0
#define RC_OFF   (BATCH * SEQL * 12)                    // 36864
#define TP_OFF   (RC_OFF + BATCH * 3 * SEQL * 4)        // 73728
#define Q_OFF    (TP_OFF + BATCH * SEQL * 4)            // 86016
#define S_OFF    (Q_OFF + BATCH * SEQL)                 // 89088
// total = 98304 floats = 384 KiB

// ---------------- Phase 0: per-frame math (tiny, VALU) ----------------
__global__ void __launch_bounds__(256)
frames_kernel(const float* __restrict__ coords,
              const unsigned char* __restrict__ mask,
              const float* __restrict__ delta_r,
              const float* __restrict__ delta_t,
              float* __restrict__ ws) {
  int idx = blockIdx.x * blockDim.x + threadIdx.x;
  if (idx >= BATCH * SEQL) return;
  int b = idx / SEQL;
  int l = idx - b * SEQL;

  const float* c = coords + (size_t)idx * 9;
  float P0 = c[0], P1 = c[1], P2 = c[2];
  float C0 = c[3], C1 = c[4], C2 = c[5];
  float N0 = c[6], N1 = c[7], N2 = c[8];

  // e1 = normalize(N - C4p)
  float v0 = N0 - C0, v1 = N1 - C1, v2 = N2 - C2;
  float nn = fmaxf(sqrtf(v0 * v0 + v1 * v1 + v2 * v2), 1e-12f);
  float e10 = v0 / nn, e11 = v1 / nn, e12 = v2 / nn;
  // e2 = normalize(u - (u.e1)e1), u = P - C4p
  float u0 = P0 - C0, u1 = P1 - C1, u2 = P2 - C2;
  float d = u0 * e10 + u1 * e11 + u2 * e12;
  float w0 = u0 - d * e10, w1 = u1 - d * e11, w2 = u2 - d * e12;
  float nw = fmaxf(sqrtf(w0 * w0 + w1 * w1 + w2 * w2), 1e-12f);
  float e20 = w0 / nw, e21 = w1 / nw, e22 = w2 / nw;
  // e3 = e1 x e2
  float e30 = e11 * e22 - e12 * e21;
  float e31 = e12 * e20 - e10 * e22;
  float e32 = e10 * e21 - e11 * e20;

  // R columns = e1,e2,e3 ; R[a][c]
  float R[3][3];
  R[0][0] = e10; R[1][0] = e11; R[2][0] = e12;
  R[0][1] = e20; R[1][1] = e21; R[2][1] = e22;
  R[0][2] = e30; R[1][2] = e31; R[2][2] = e32;

  bool mk = mask[idx] != 0;
  if (!mk) {
    for (int a = 0; a < 3; ++a)
      for (int cc = 0; cc < 3; ++cc) R[a][cc] = (a == cc) ? 1.f : 0.f;
  }
  float mf = mk ? 1.f : 0.f;

  float dr0 = delta_r[(size_t)idx * 3 + 0] * mf;
  float dr1 = delta_r[(size_t)idx * 3 + 1] * mf;
  float dr2 = delta_r[(size_t)idx * 3 + 2] * mf;
  float dtx = delta_t[(size_t)idx * 3 + 0] * mf;
  float dty = delta_t[(size_t)idx * 3 + 1] * mf;
  float dtz = delta_t[(size_t)idx * 3 + 2] * mf;

  float ang = fmaxf(sqrtf(dr0 * dr0 + dr1 * dr1 + dr2 * dr2), 1e-8f);
  float ax = dr0 / ang, ay = dr1 / ang, az = dr2 / ang;
  float ca = cosf(ang), sa = sinf(ang), oc = 1.f - ca;

  // Rodrigues: R_delta = ca*I + sa*K + oc*axis axis^T,  K = skew(axis)
  float Rd[3][3];
  Rd[0][0] = ca + oc * ax * ax;       Rd[0][1] = -sa * az + oc * ax * ay; Rd[0][2] = sa * ay + oc * ax * az;
  Rd[1][0] = sa * az + oc * ay * ax;  Rd[1][1] = ca + oc * ay * ay;       Rd[1][2] = -sa * ax + oc * ay * az;
  Rd[2][0] = -sa * ay + oc * az * ax; Rd[2][1] = sa * ax + oc * az * ay;  Rd[2][2] = ca + oc * az * az;

  float Rn[3][3];
  for (int i = 0; i < 3; ++i)
    for (int j = 0; j < 3; ++j)
      Rn[i][j] = Rd[i][0] * R[0][j] + Rd[i][1] * R[1][j] + Rd[i][2] * R[2][j];

  // t_new = C4p + R @ delta_t  (note: old R)
  float tn0 = C0 + R[0][0] * dtx + R[0][1] * dty + R[0][2] * dtz;
  float tn1 = C1 + R[1][0] * dtx + R[1][1] * dty + R[1][2] * dtz;
  float tn2 = C2 + R[2][0] * dtx + R[2][1] * dty + R[2][2] * dtz;

  // ---- emit fragment-friendly tables ----
  float* rv = ws + RV12_OFF + (size_t)idx * 12;
  for (int a = 0; a < 3; ++a)
    for (int cc = 0; cc < 3; ++cc) rv[a * 3 + cc] = Rn[a][cc];
  rv[9] = 0.f; rv[10] = 0.f; rv[11] = 0.f;

  for (int cc = 0; cc < 3; ++cc) {
    float* rc = ws + RC_OFF + (((size_t)b * 3 + cc) * SEQL + l) * 4;
    rc[0] = Rn[0][cc]; rc[1] = Rn[1][cc]; rc[2] = Rn[2][cc]; rc[3] = 0.f;
  }
  float* tp = ws + TP_OFF + (size_t)idx * 4;
  tp[0] = tn0; tp[1] = tn1; tp[2] = tn2; tp[3] = 0.f;

  ws[Q_OFF + idx] = tn0 * tn0 + tn1 * tn1 + tn2 * tn2;

  float* sp = ws + S_OFF + (size_t)idx * 3;
  for (int cc = 0; cc < 3; ++cc)
    sp[cc] = Rn[0][cc] * tn0 + Rn[1][cc] * tn1 + Rn[2][cc] * tn2;
}

// ---------------- Phase 1: pairwise invariants via WMMA ----------------
__device__ __forceinline__ v8f wmma4(v2f a, v2f b, v8f c) {
  // (neg_a, A, neg_b, B, c_mod, C, reuse_a, reuse_b)
  return __builtin_amdgcn_wmma_f32_16x16x4_f32(false, a, false, b, (short)0, c,
                                               false, false);
}

// LDS tile: 16 rows, each 16 pairs * 7 ch = 112 dwords, padded to 116 so the
// two half-wave store patterns (bank = 7*m vs 7*m + 32 mod 64) never collide.
#define LDSROW 116

__global__ void __launch_bounds__(32)
invariants_kernel(const float* __restrict__ ws, float* __restrict__ out) {
  __shared__ __align__(16) float tile[16 * LDSROW];

  const int b  = blockIdx.z;
  const int i0 = blockIdx.x * 16;
  const int j0 = blockIdx.y * 16;
  const int lane = threadIdx.x;
  const int m = lane & 15;
  const int h = lane >> 4;     // K-half selector for A/B fragments

  const float* Rv = ws + RV12_OFF + (size_t)b * SEQL * 12;
  const float* Rc = ws + RC_OFF + (size_t)b * 3 * SEQL * 4;
  const float* Tp = ws + TP_OFF + (size_t)b * SEQL * 4;
  const float* qb = ws + Q_OFF + (size_t)b * SEQL;
  const float* sb = ws + S_OFF + (size_t)b * SEQL * 3;

  // A-fragments (rows i0..i0+15) and B-fragments (cols j0..j0+15) share the
  // same per-lane gather: element [row m][K = kc + 2h .. +1] as one b64 load.
  const float* rvi = Rv + (size_t)(i0 + m) * 12 + 2 * h;
  const float* rvj = Rv + (size_t)(j0 + m) * 12 + 2 * h;
  v2f aT0 = *(const v2f*)(rvi + 0);
  v2f aT1 = *(const v2f*)(rvi + 4);
  v2f aT2 = *(const v2f*)(rvi + 8);
  v2f bT0 = *(const v2f*)(rvj + 0);
  v2f bT1 = *(const v2f*)(rvj + 4);
  v2f bT2 = *(const v2f*)(rvj + 8);

  v2f tA = *(const v2f*)(Tp + (size_t)(i0 + m) * 4 + 2 * h);
  v2f tB = *(const v2f*)(Tp + (size_t)(j0 + m) * 4 + 2 * h);

  v2f a0 = *(const v2f*)(Rc + ((size_t)0 * SEQL + i0 + m) * 4 + 2 * h);
  v2f a1 = *(const v2f*)(Rc + ((size_t)1 * SEQL + i0 + m) * 4 + 2 * h);
  v2f a2 = *(const v2f*)(Rc + ((size_t)2 * SEQL + i0 + m) * 4 + 2 * h);

  // trace(R_i^T R_j): K=9 padded to 12 -> 3 WMMAs
  v8f tr = {};
  tr = wmma4(aT0, bT0, tr);
  tr = wmma4(aT1, bT1, tr);
  tr = wmma4(aT2, bT2, tr);
  // t_i . t_j : K=3 padded to 4
  v8f td = {};
  td = wmma4(tA, tB, td);
  // (R_i^T t_j)_c : three K=3 GEMMs sharing the t_j B fragment
  v8f p0 = {}, p1 = {}, p2 = {};
  p0 = wmma4(a0, tB, p0);
  p1 = wmma4(a1, tB, p1);
  p2 = wmma4(a2, tB, p2);

  const float qj = qb[j0 + m];

#pragma unroll
  for (int r = 0; r < 8; ++r) {
    const int row = r + 8 * h;           // C/D layout: VGPR r holds rows r, r+8
    const int i = i0 + row;
    const float qi = qb[i];
    const float s0 = sb[(size_t)i * 3 + 0];
    const float s1 = sb[(size_t)i * 3 + 1];
    const float s2 = sb[(size_t)i * 3 + 2];

    const float trv = tr[r];
    // dist = max(||dt||,1e-4) = sqrt(max(d2,1e-8)); log(dist) = 0.5*log(d2c)
    const float d2 = qi + qj - 2.f * td[r];
    const float d2c = fmaxf(d2, 1e-8f);
    const float dist = __builtin_amdgcn_sqrtf(d2c);
    const float logd = 0.5f * __logf(d2c);
    const float c0 = p0[r] - s0;
    const float c1 = p1[r] - s1;
    const float c2 = p2[r] - s2;
    const float cosa = fminf(fmaxf((trv - 1.f) * 0.5f, -1.f), 1.f);

    float* o = tile + row * LDSROW + m * 7;
    o[0] = dist;
    o[1] = c0;
    o[2] = c1;
    o[3] = c2;
    o[4] = trv;
    o[5] = cosa;
    o[6] = logd;
  }

  // Drain the tile with fully coalesced 16B stores: each output row is a
  // contiguous, 16B-aligned 448B run (28 float4 chunks), 448 chunks total.
  // Single wave per workgroup: LDS RAW ordered by DScnt (compiler-inserted).
#pragma unroll
  for (int k = 0; k < 14; ++k) {
    const int t = lane + 32 * k;         // 0..447
    const int row = t / 28;
    const int c = t - row * 28;
    const v4f val = *(const v4f*)(tile + row * LDSROW + c * 4);
    float* g = out + (((size_t)b * SEQL + i0 + row) * SEQL + j0) * 7 + c * 4;
    *(v4f*)g = val;
  }
}

extern "C" void kernel_launch(void* const* d_in, const int* in_sizes, int n_in,
                              void* d_out, int out_size, void* d_ws,
                              size_t ws_size, hipStream_t stream) {
  const float* coords = (const float*)d_in[0];
  const unsigned char* mask = (const unsigned char*)d_in[1];  // numpy bool: 1B
  const float* delta_r = (const float*)d_in[2];
  const float* delta_t = (const float*)d_in[3];
  float* ws = (float*)d_ws;   // needs 384 KiB
  float* out = (float*)d_out; // (B, L, L, 7) f32

  dim3 g0((BATCH * SEQL + 255) / 256);
  frames_kernel<<<g0, dim3(256), 0, stream>>>(coords, mask, delta_r, delta_t,
                                              ws);

  dim3 g1(NTILE, NTILE, BATCH);
  invariants_kernel<<<g1, dim3(32), 0, stream>>>(ws, out);
}